// Encoder_48550310314044
// MI455X (gfx1250) — compile-verified
//
#include <hip/hip_runtime.h>

// GCN 2-layer: h = relu(Agg(X@W1)+b1); mu = Agg(h@W2)+b2
// Agg = D^-1/2 (A+I) D^-1/2 scatter-add over edges.
// MI455X: scatter-bound (~2.6 GB irregular traffic @ 23.3 TB/s ~ 110us floor).
// H buffers (51MB) are L2-resident (192MB L2) so per-edge gathers hit L2.
// GEMMs (5 GFLOP) use fp32 WMMA (V_WMMA_F32_16X16X4_F32) for exact fp32 math;
// B matrix staged in LDS (padded, conflict-free) so the WMMA inner loop feeds
// from ds_load instead of re-streaming L2.

typedef __attribute__((ext_vector_type(2))) float v2f;
typedef __attribute__((ext_vector_type(8))) float v8f;

#define INC  128
#define HIDC 128
#define OUTC 64

static inline int ceilDiv(int a, int b) { return (a + b - 1) / b; }

// ---------------- degree / normalization ----------------

__global__ void deg_init(float* __restrict__ deg, int n) {
    int i = blockIdx.x * blockDim.x + threadIdx.x;
    if (i < n) deg[i] = 1.0f;  // self-loop contribution
}

__global__ void deg_accum(const int* __restrict__ col, float* __restrict__ deg, int e) {
    int i = blockIdx.x * blockDim.x + threadIdx.x;
    if (i < e) atomicAdd(&deg[col[i]], 1.0f);
}

__global__ void deg_rsqrt(float* __restrict__ deg, int n) {
    int i = blockIdx.x * blockDim.x + threadIdx.x;
    if (i < n) deg[i] = rsqrtf(deg[i]);  // deg >= 1 always (self loops)
}

// ---------------- WMMA fp32 GEMM: C[M x OUTW] = A[M x 128] * B[128 x OUTW] ----------------
// One wave computes a 16-row x OUTW tile. K advances 4 per V_WMMA_F32_16X16X4_F32.
// A 16x4 f32 layout (ISA 7.12.2): lanes 0-15 hold {K=0,K=1}, lanes 16-31 hold {K=2,K=3},
// M = lane & 15. B 4x16 mirrored: N = lane & 15, VGPRs hold {K,K+1} per half-wave.
// C/D 16x16 f32: VGPR v -> M = v + 8*(lane>=16), N = lane & 15.
// B is staged in LDS once per block (8 waves share). Row pitch OUTW+16 floats:
// half-wave row delta 2*(OUTW+16) % 64 == 32 banks -> the two half-wave access
// sets {t*16..t*16+15} and {t*16+32..} are disjoint mod 64 -> conflict-free.

template <int OUTW>
__global__ void gemm_wmma_f32(const float* __restrict__ A,
                              const float* __restrict__ B,
                              float* __restrict__ C, int M) {
    constexpr int PAD   = 16;
    constexpr int PITCH = OUTW + PAD;
    __shared__ float Bs[128 * PITCH];

    // cooperative stage of B (128 x OUTW) into LDS, float4 per thread
    constexpr int V4PerRow = OUTW / 4;
    for (int i = threadIdx.x; i < 128 * V4PerRow; i += blockDim.x) {
        const int rr = i / V4PerRow;
        const int cc = (i - rr * V4PerRow) * 4;
        *(float4*)&Bs[rr * PITCH + cc] = *(const float4*)&B[rr * OUTW + cc];
    }
    __syncthreads();

    const int lane = threadIdx.x & 31;
    const int wave = threadIdx.x >> 5;
    const int wavesPerBlock = blockDim.x >> 5;
    const int tile = blockIdx.x * wavesPerBlock + wave;   // 16-row tile index
    const int row0 = tile * 16;
    if (row0 >= M) return;  // after barrier; wave-uniform -> EXEC all-ones for WMMA

    const int half = lane >> 4;   // 0 or 1
    const int l16  = lane & 15;

    constexpr int NT = OUTW / 16;
    v8f acc[NT] = {};

    const float* arow = A + (size_t)(row0 + l16) * 128 + half * 2;

    for (int k = 0; k < 128; k += 4) {
        v2f a;
        a.x = arow[k];
        a.y = arow[k + 1];
        const int kb = k + half * 2;
        const float* brow0 = &Bs[kb * PITCH];
        const float* brow1 = &Bs[(kb + 1) * PITCH];
#pragma unroll
        for (int t = 0; t < NT; ++t) {
            const int colb = t * 16 + l16;
            v2f b;
            b.x = brow0[colb];
            b.y = brow1[colb];
            acc[t] = __builtin_amdgcn_wmma_f32_16x16x4_f32(
                false, a, false, b, (short)0, acc[t], false, false);
        }
    }

#pragma unroll
    for (int t = 0; t < NT; ++t) {
#pragma unroll
        for (int v = 0; v < 8; ++v) {
            const int r = row0 + v + half * 8;
            const int c = t * 16 + l16;
            C[(size_t)r * OUTW + c] = acc[t][v];
        }
    }
}

// ---------------- edge scatter-add: agg[col] += H[row] * dinv[row]*dinv[col] ----------------
// One wave32 per (edge or self-loop); each lane moves W/32 contiguous floats.

template <int W>
__global__ void scatter_edges(const int* __restrict__ ei,      // [2, E] flat
                              const float* __restrict__ dinv,  // [n]
                              const float* __restrict__ H,     // [n, W]
                              float* __restrict__ agg,         // [n, W]
                              int e, int n) {
    const int wave = (int)((blockIdx.x * blockDim.x + threadIdx.x) >> 5);
    const int lane = threadIdx.x & 31;
    const int total = e + n;
    if (wave >= total) return;

    int r, c;
    if (wave < e) {
        r = ei[wave];      // edge_index[0][wave] (source)
        c = ei[e + wave];  // edge_index[1][wave] (target)
    } else {
        r = c = wave - e;  // self loop
    }
    const float nrm = dinv[r] * dinv[c];

    constexpr int PER = W / 32;  // 4 (W=128) or 2 (W=64)
    const float* src = H + (size_t)r * W + lane * PER;
    float* dst = agg + (size_t)c * W + lane * PER;
#pragma unroll
    for (int i = 0; i < PER; ++i) {
        atomicAdd(&dst[i], src[i] * nrm);
    }
}

// ---------------- bias (+ optional relu) ----------------

template <bool RELU, int MASK>
__global__ void bias_act(float* __restrict__ h, const float* __restrict__ b, int total) {
    int i = blockIdx.x * blockDim.x + threadIdx.x;
    if (i < total) {
        float v = h[i] + b[i & MASK];
        if (RELU) v = v > 0.0f ? v : 0.0f;
        h[i] = v;
    }
}

// ---------------- launch ----------------

extern "C" void kernel_launch(void* const* d_in, const int* in_sizes, int n_in,
                              void* d_out, int out_size, void* d_ws, size_t ws_size,
                              hipStream_t stream) {
    const float* x  = (const float*)d_in[0];
    const int*   ei = (const int*)d_in[1];
    const float* W1 = (const float*)d_in[2];
    const float* b1 = (const float*)d_in[3];
    const float* W2 = (const float*)d_in[4];
    const float* b2 = (const float*)d_in[5];
    float* out = (float*)d_out;

    const int n = in_sizes[0] / INC;  // 100000
    const int e = in_sizes[1] / 2;    // 1600000

    // workspace: dinv[n] | buf1[n*128] | buf2[n*128]
    char* ws = (char*)d_ws;
    size_t off = 0;
    float* dinv = (float*)(ws + off); off += (((size_t)n * 4) + 511) & ~(size_t)511;
    float* buf1 = (float*)(ws + off); off += (size_t)n * HIDC * 4;   // Ht, later H2
    float* buf2 = (float*)(ws + off); off += (size_t)n * HIDC * 4;   // agg1 -> H

    hipMemsetAsync(buf2, 0, (size_t)n * HIDC * 4, stream);  // agg1 = 0
    hipMemsetAsync(out, 0, (size_t)n * OUTC * 4, stream);   // agg2 = 0

    // normalization
    deg_init<<<ceilDiv(n, 256), 256, 0, stream>>>(dinv, n);
    deg_accum<<<ceilDiv(e, 256), 256, 0, stream>>>(ei + e, dinv, e);
    deg_rsqrt<<<ceilDiv(n, 256), 256, 0, stream>>>(dinv, n);

    const int tiles = ceilDiv(n, 16);          // 6250
    const int totalEdges = e + n;              // edges + self loops

    // layer 1: Ht = X @ W1
    gemm_wmma_f32<HIDC><<<ceilDiv(tiles, 8), 256, 0, stream>>>(x, W1, buf1, n);
    // agg1 += norm * Ht[row] at col
    scatter_edges<HIDC><<<ceilDiv(totalEdges, 8), 256, 0, stream>>>(ei, dinv, buf1, buf2, e, n);
    // H = relu(agg1 + b1)
    bias_act<true, HIDC - 1><<<ceilDiv(n * HIDC, 256), 256, 0, stream>>>(buf2, b1, n * HIDC);

    // layer 2: H2 = H @ W2 (reuse buf1)
    gemm_wmma_f32<OUTC><<<ceilDiv(tiles, 8), 256, 0, stream>>>(buf2, W2, buf1, n);
    // out += norm * H2[row] at col
    scatter_edges<OUTC><<<ceilDiv(totalEdges, 8), 256, 0, stream>>>(ei, dinv, buf1, out, e, n);
    // out += b2
    bias_act<false, OUTC - 1><<<ceilDiv(n * OUTC, 256), 256, 0, stream>>>(out, b2, n * OUTC);
}